// DualAttentionEEGNet_41918880809096
// MI455X (gfx1250) — compile-verified
//
#include <hip/hip_runtime.h>
#include <math.h>

#define C_CH 64
#define NBANDS 5
#define DB 16
#define DC 64
#define NCLS 4
#define LN_EPS 1e-5f
#define HB_PITCH 20     // 80-byte rows: 16B-aligned for b128 LDS, conflict-free
#define HCH_PITCH 66

typedef float v2f __attribute__((ext_vector_type(2)));
typedef float v4f __attribute__((ext_vector_type(4)));
typedef float v8f __attribute__((ext_vector_type(8)));

__device__ __forceinline__ int   biti(float f) { return __builtin_bit_cast(int, f); }
__device__ __forceinline__ float bitf(int i)   { return __builtin_bit_cast(float, i); }

// Raw hardware reciprocal (V_RCP_F32); ~1 ulp, plenty for the erf poly below.
__device__ __forceinline__ float rcp_hw(float x) {
#if __has_builtin(__builtin_amdgcn_rcpf)
    return __builtin_amdgcn_rcpf(x);
#else
    return 1.0f / x;
#endif
}

// Branch-free minimax erf (Abramowitz-Stegun 7.1.26, |err| <= 1.5e-7).
// ~10 VALU ops + v_rcp_f32 + v_exp_f32, no SALU divergence.
__device__ __forceinline__ float erf_fast(float z) {
    float az = fabsf(z);
    float t  = rcp_hw(fmaf(0.3275911f, az, 1.0f));
    float p  = fmaf(1.061405429f, t, -1.453152027f);
    p = fmaf(p, t, 1.421413741f);
    p = fmaf(p, t, -0.284496736f);
    p = fmaf(p, t, 0.254829592f);
    p *= t;
    float e = __expf(-az * az);          // hardware v_exp_f32
    float r = fmaf(-p, e, 1.0f);
    return copysignf(r, z);
}

__device__ __forceinline__ float gelu_erf(float x) {
    return 0.5f * x * (1.0f + erf_fast(x * 0.70710678118654752440f));
}

// Hardware tanh (CDNA5 V_TANH_F32 trans op)
__device__ __forceinline__ float tanh_hw(float x) {
#if __has_builtin(__builtin_amdgcn_tanhf)
    return __builtin_amdgcn_tanhf(x);
#else
    float r;
    asm volatile("v_tanh_f32 %0, %1\n\tv_nop" : "=v"(r) : "v"(x));
    return r;
#endif
}

// ---- cross-lane primitives: DPP/permlane instead of ds_bpermute round-trips
template <int XM>   // XM in {1,2,4}: xor within each 8-lane group via DPP8
__device__ __forceinline__ float dpp8_xor_f(float x) {
#if __has_builtin(__builtin_amdgcn_mov_dpp8)
    constexpr int s = ((0 ^ XM) << 0)  | ((1 ^ XM) << 3)  | ((2 ^ XM) << 6)  |
                      ((3 ^ XM) << 9)  | ((4 ^ XM) << 12) | ((5 ^ XM) << 15) |
                      ((6 ^ XM) << 18) | ((7 ^ XM) << 21);
    return bitf(__builtin_amdgcn_mov_dpp8(biti(x), s));
#else
    return __shfl_xor(x, XM, 32);
#endif
}
__device__ __forceinline__ float dpp_xor8_f(float x) {   // i^8 == row_ror:8 in a 16-lane row
#if __has_builtin(__builtin_amdgcn_update_dpp)
    int xi = biti(x);
    return bitf(__builtin_amdgcn_update_dpp(xi, xi, 0x128, 0xf, 0xf, true));
#else
    return __shfl_xor(x, 8, 32);
#endif
}
__device__ __forceinline__ float dpp_xor16_f(float x) {  // swap 16-lane halves
#if __has_builtin(__builtin_amdgcn_permlanex16)
    int xi = biti(x);
    return bitf(__builtin_amdgcn_permlanex16(xi, xi, 0x76543210, 0xfedcba98, false, false));
#else
    return __shfl_xor(x, 16, 32);
#endif
}

__device__ __forceinline__ float red8_add(float v) {     // sum over 8-lane group
    v += dpp8_xor_f<1>(v); v += dpp8_xor_f<2>(v); v += dpp8_xor_f<4>(v);
    return v;
}
__device__ __forceinline__ float red16_add(float v) {    // sum over 16-lane row
    v = red8_add(v); v += dpp_xor8_f(v); return v;
}
__device__ __forceinline__ float red32_add(float v) {    // sum over wave
    v = red16_add(v); v += dpp_xor16_f(v); return v;
}
__device__ __forceinline__ float red32_max(float v) {
    v = fmaxf(v, dpp8_xor_f<1>(v)); v = fmaxf(v, dpp8_xor_f<2>(v));
    v = fmaxf(v, dpp8_xor_f<4>(v)); v = fmaxf(v, dpp_xor8_f(v));
    v = fmaxf(v, dpp_xor16_f(v));   return v;
}

__global__ __launch_bounds__(64)
void dual_attn_eeg_kernel(
    const float* __restrict__ x, const int* __restrict__ channel_mask,
    const float* __restrict__ W_be, const float* __restrict__ b_be,
    const float* __restrict__ W_ba, const float* __restrict__ b_ba,
    const float* __restrict__ v_ba,
    const float* __restrict__ W_cp, const float* __restrict__ b_cp,
    const float* __restrict__ g_ln1, const float* __restrict__ be_ln1,
    const float* __restrict__ W_ca, const float* __restrict__ b_ca,
    const float* __restrict__ v_ca,
    const float* __restrict__ g_ln2, const float* __restrict__ be_ln2,
    const float* __restrict__ W_c1, const float* __restrict__ b_c1,
    const float* __restrict__ W_c2, const float* __restrict__ b_c2,
    float* __restrict__ out_logits, float* __restrict__ out_alpha,
    float* __restrict__ out_beta)
{
    const int b    = blockIdx.x;
    const int t    = threadIdx.x;      // 0..63; thread == channel
    const int lane = t & 31;
    const int wv   = t >> 5;           // wave id (0,1)
    const int ln16 = lane & 15;
    const int hi   = (lane >> 4) & 1;  // 16-lane half of the wave

    // hb_s (320 x 20 floats) dead after band pooling; hch_s (64 x 66) written
    // only after the barrier that follows pooling -> union the buffers.
    __shared__ __align__(16) float ub_s[(C_CH * NBANDS) * HB_PITCH];   // 6400 floats
    float (*hb_s)[HB_PITCH]   = (float (*)[HB_PITCH])ub_s;             // [320][20]
    float (*hch_s)[HCH_PITCH] = (float (*)[HCH_PITCH])ub_s;            // [64][66]

    __shared__ __align__(16) float hc_s[C_CH][HB_PITCH];               // pooled band feats
    __shared__ float eband_s[C_CH * NBANDS];
    __shared__ float e_s[C_CH];
    __shared__ float al_s[C_CH];
    __shared__ float ctx_s[DC];
    __shared__ float h1_s[DC];
    __shared__ float red_s[4];

    const v8f vzero = {0.f, 0.f, 0.f, 0.f, 0.f, 0.f, 0.f, 0.f};

    // ------------------------------------------------------------------
    // Stage 1a: band embedding (erf-GELU), one channel per lane -> LDS (b128).
    // ------------------------------------------------------------------
    {
        const int c = t;
        const float* xr = x + ((size_t)b * C_CH + c) * NBANDS;
#pragma unroll
        for (int n = 0; n < NBANDS; ++n) {
            float xv = xr[n];
#pragma unroll
            for (int d4 = 0; d4 < DB / 4; ++d4) {
                v4f h;
#pragma unroll
                for (int u = 0; u < 4; ++u)
                    h[u] = gelu_erf(xv * W_be[4 * d4 + u] + b_be[4 * d4 + u]);
                *(v4f*)&hb_s[c * NBANDS + n][4 * d4] = h;
            }
        }
    }

    // ------------------------------------------------------------------
    // Stage 1b: band attention GEMM (320x16)@(16x8) via WMMA.
    // Each wave owns 10 M-tiles = its own lanes' rows (intra-wave LDS order).
    // ------------------------------------------------------------------
    {
        v2f bba[4];                    // W_ba fragments, zero-padded N=8->16
#pragma unroll
        for (int k = 0; k < 4; ++k) {
            int kk = 4 * k + 2 * hi;
            bba[k].x = (ln16 < 8) ? W_ba[kk * 8 + ln16] : 0.0f;
            bba[k].y = (ln16 < 8) ? W_ba[(kk + 1) * 8 + ln16] : 0.0f;
        }
        float bbias = (ln16 < 8) ? b_ba[ln16] : 0.0f;
        float vba   = (ln16 < 8) ? v_ba[ln16] : 0.0f;

        for (int mt = 10 * wv; mt < 10 * wv + 10; ++mt) {
            const int r0 = mt * 16;
            v8f acc = vzero;
#pragma unroll
            for (int k = 0; k < 4; ++k) {
                int kk = 4 * k + 2 * hi;
                v2f a;
                a.x = hb_s[r0 + ln16][kk];
                a.y = hb_s[r0 + ln16][kk + 1];
                acc = __builtin_amdgcn_wmma_f32_16x16x4_f32(
                    false, a, false, bba[k], (short)0, acc, false, false);
            }
            // e = v_ba . tanh(row + b_ba): columns 0..7 only -> 3-step DPP8 sum
            float ssel = 0.0f;
#pragma unroll
            for (int v = 0; v < 8; ++v) {
                float s = tanh_hw(acc[v] + bbias) * vba;   // 0 in cols >= 8
                s = red8_add(s);
                ssel = (ln16 == v) ? s : ssel;
            }
            if (ln16 < 8) eband_s[r0 + ln16 + 8 * hi] = ssel;
        }
    }

    // ------------------------------------------------------------------
    // Stage 1c: per-channel band softmax + weighted pool (lane = channel).
    // ------------------------------------------------------------------
    {
        const int c = t;
        float e[NBANDS];
#pragma unroll
        for (int n = 0; n < NBANDS; ++n) e[n] = eband_s[c * NBANDS + n];
        float mx = e[0];
#pragma unroll
        for (int n = 1; n < NBANDS; ++n) mx = fmaxf(mx, e[n]);
        float den = 0.0f, bt[NBANDS];
#pragma unroll
        for (int n = 0; n < NBANDS; ++n) { bt[n] = __expf(e[n] - mx); den += bt[n]; }
        float inv = rcp_hw(den);
        float* ob = out_beta + ((size_t)b * C_CH + c) * NBANDS;
#pragma unroll
        for (int n = 0; n < NBANDS; ++n) { bt[n] *= inv; ob[n] = bt[n]; }

        v4f hcv[DB / 4];
#pragma unroll
        for (int d4 = 0; d4 < DB / 4; ++d4) hcv[d4] = (v4f){0.f, 0.f, 0.f, 0.f};
#pragma unroll
        for (int n = 0; n < NBANDS; ++n) {
            float w = bt[n];
#pragma unroll
            for (int d4 = 0; d4 < DB / 4; ++d4)
                hcv[d4] += w * *(v4f*)&hb_s[c * NBANDS + n][4 * d4];
        }
#pragma unroll
        for (int d4 = 0; d4 < DB / 4; ++d4)
            *(v4f*)&hc_s[c][4 * d4] = hcv[d4];
    }
    __syncthreads();   // hb_s dead past here; hch_s may reuse the buffer

    // ------------------------------------------------------------------
    // Stage 2: channel-proj GEMM + LN + GELU, then channel-attn GEMM.
    // ------------------------------------------------------------------
    for (int mt = 2 * wv; mt < 2 * wv + 2; ++mt) {
        const int c0 = mt * 16;

        v2f afr[4];
#pragma unroll
        for (int k = 0; k < 4; ++k) {
            int kk = 4 * k + 2 * hi;
            afr[k].x = hc_s[c0 + ln16][kk];
            afr[k].y = hc_s[c0 + ln16][kk + 1];
        }

        v8f acc[4];
#pragma unroll
        for (int j = 0; j < 4; ++j) acc[j] = vzero;
#pragma unroll
        for (int j = 0; j < 4; ++j) {
            int n = 16 * j + ln16;
#pragma unroll
            for (int k = 0; k < 4; ++k) {
                int kk = 4 * k + 2 * hi;
                v2f bfr;
                bfr.x = W_cp[kk * DC + n];
                bfr.y = W_cp[(kk + 1) * DC + n];
                acc[j] = __builtin_amdgcn_wmma_f32_16x16x4_f32(
                    false, afr[k], false, bfr, (short)0, acc[j], false, false);
            }
        }

        // bias + LayerNorm(64) + GELU on the D-matrix layout
        float vals[4][8];
#pragma unroll
        for (int j = 0; j < 4; ++j) {
            float bias = b_cp[16 * j + ln16];
#pragma unroll
            for (int v = 0; v < 8; ++v) vals[j][v] = acc[j][v] + bias;
        }
#pragma unroll
        for (int v = 0; v < 8; ++v) {
            float s = 0.0f, q = 0.0f;
#pragma unroll
            for (int j = 0; j < 4; ++j) { s += vals[j][v]; q += vals[j][v] * vals[j][v]; }
            s = red16_add(s);
            q = red16_add(q);
            float mu   = s * (1.0f / 64.0f);
            float var  = q * (1.0f / 64.0f) - mu * mu;
            float rstd = rsqrtf(var + LN_EPS);
            int   r    = c0 + v + 8 * hi;
#pragma unroll
            for (int j = 0; j < 4; ++j) {
                int n = 16 * j + ln16;
                float y = (vals[j][v] - mu) * rstd * g_ln1[n] + be_ln1[n];
                hch_s[r][n] = gelu_erf(y);
            }
        }

        // channel-attention GEMM: K=64 -> 16 WMMA K-steps, N=32 -> 2 tiles
        v8f acc2[2];
        acc2[0] = vzero; acc2[1] = vzero;
#pragma unroll
        for (int k = 0; k < 16; ++k) {
            int kk = 4 * k + 2 * hi;
            v2f a2;
            a2.x = hch_s[c0 + ln16][kk];
            a2.y = hch_s[c0 + ln16][kk + 1];
#pragma unroll
            for (int j = 0; j < 2; ++j) {
                int n = 16 * j + ln16;
                v2f b2;
                b2.x = W_ca[kk * (DC / 2) + n];
                b2.y = W_ca[(kk + 1) * (DC / 2) + n];
                acc2[j] = __builtin_amdgcn_wmma_f32_16x16x4_f32(
                    false, a2, false, b2, (short)0, acc2[j], false, false);
            }
        }
        float ssel = 0.0f;
#pragma unroll
        for (int v = 0; v < 8; ++v) {
            float s = 0.0f;
#pragma unroll
            for (int j = 0; j < 2; ++j) {
                int n = 16 * j + ln16;
                s += tanh_hw(acc2[j][v] + b_ca[n]) * v_ca[n];
            }
            s = red16_add(s);
            ssel = (ln16 == v) ? s : ssel;
        }
        if (ln16 < 8) e_s[c0 + ln16 + 8 * hi] = ssel;
    }
    __syncthreads();

    // ------------------------------------------------------------------
    // Stage 3: masked softmax over the 64 channels
    // ------------------------------------------------------------------
    float ec = e_s[t];
    if (channel_mask[(size_t)b * C_CH + t] == 0) ec = -__builtin_inff();
    float m = red32_max(ec);
    if (lane == 0) red_s[wv] = m;
    __syncthreads();
    m = fmaxf(red_s[0], red_s[1]);
    __syncthreads();
    float ex = __expf(ec - m);
    float sden = red32_add(ex);
    if (lane == 0) red_s[wv] = sden;
    __syncthreads();
    sden = red_s[0] + red_s[1];
    __syncthreads();
    float alpha = ex / sden;
    al_s[t] = alpha;
    out_alpha[(size_t)b * C_CH + t] = alpha;
    __syncthreads();

    // ------------------------------------------------------------------
    // Stage 4: context pool + LayerNorm2 + classifier
    // ------------------------------------------------------------------
    float cx = 0.0f;
#pragma unroll 8
    for (int c = 0; c < C_CH; ++c) cx += al_s[c] * hch_s[c][t];

    float ssum = red32_add(cx);
    float qsum = red32_add(cx * cx);
    if (lane == 0) { red_s[wv] = ssum; red_s[2 + wv] = qsum; }
    __syncthreads();
    float mu  = (red_s[0] + red_s[1]) * (1.0f / 64.0f);
    float var = (red_s[2] + red_s[3]) * (1.0f / 64.0f) - mu * mu;
    float lncx = (cx - mu) * rsqrtf(var + LN_EPS) * g_ln2[t] + be_ln2[t];
    ctx_s[t] = lncx;
    __syncthreads();

    float h = b_c1[t];
#pragma unroll 8
    for (int k = 0; k < DC; ++k) h += ctx_s[k] * W_c1[k * DC + t];
    h1_s[t] = gelu_erf(h);
    __syncthreads();

    if (t < NCLS) {
        float lg = b_c2[t];
#pragma unroll 8
        for (int k = 0; k < DC; ++k) lg += h1_s[k] * W_c2[k * NCLS + t];
        out_logits[(size_t)b * NCLS + t] = lg;
    }
}

extern "C" void kernel_launch(void* const* d_in, const int* in_sizes, int n_in,
                              void* d_out, int out_size, void* d_ws, size_t ws_size,
                              hipStream_t stream) {
    const float* x      = (const float*)d_in[0];
    const int*   cmask  = (const int*)  d_in[1];
    const float* W_be   = (const float*)d_in[2];
    const float* b_be   = (const float*)d_in[3];
    const float* W_ba   = (const float*)d_in[4];
    const float* b_ba   = (const float*)d_in[5];
    const float* v_ba   = (const float*)d_in[6];
    const float* W_cp   = (const float*)d_in[7];
    const float* b_cp   = (const float*)d_in[8];
    const float* g_ln1  = (const float*)d_in[9];
    const float* be_ln1 = (const float*)d_in[10];
    const float* W_ca   = (const float*)d_in[11];
    const float* b_ca   = (const float*)d_in[12];
    const float* v_ca   = (const float*)d_in[13];
    const float* g_ln2  = (const float*)d_in[14];
    const float* be_ln2 = (const float*)d_in[15];
    const float* W_c1   = (const float*)d_in[16];
    const float* b_c1   = (const float*)d_in[17];
    const float* W_c2   = (const float*)d_in[18];
    const float* b_c2   = (const float*)d_in[19];

    const int B = in_sizes[0] / (C_CH * NBANDS);   // 16384

    float* out        = (float*)d_out;
    float* out_logits = out;                                   // (B, 4)
    float* out_alpha  = out + (size_t)B * NCLS;                // (B, 64)
    float* out_beta   = out_alpha + (size_t)B * C_CH;          // (B, 64, 5)

    dual_attn_eeg_kernel<<<B, 64, 0, stream>>>(
        x, cmask, W_be, b_be, W_ba, b_ba, v_ba, W_cp, b_cp,
        g_ln1, be_ln1, W_ca, b_ca, v_ca, g_ln2, be_ln2,
        W_c1, b_c1, W_c2, b_c2,
        out_logits, out_alpha, out_beta);
}